// Patch2Im_20461224198690
// MI455X (gfx1250) — compile-verified
//
#include <hip/hip_runtime.h>
#include <stdint.h>
#include <stddef.h>

#define AS_GLOBAL __attribute__((address_space(1)))
#define AS_LDS    __attribute__((address_space(3)))

// ---------------- problem constants -----------------------------------------------------
// x_patch: [B=4, C=64, K=7, K=7, nH=85, nW=85] f32
// fold stride 3  ->  full image 259x259, crop 3 each side -> out [4,64,253,253] f32
constexpr int KK        = 7;
constexpr int NH        = 85;
constexpr int NW        = 85;
constexpr int PADT      = 3;
constexpr int HO        = 253;
constexpr int WO        = 253;
constexpr int PLANE     = NH * NW;        // 7225
constexpr int BC        = 4 * 64;         // 256
constexpr int SEG_LEN   = NW;             // 85 floats per staged plane-row
constexpr int SEG_STRIDE= 88;             // padded LDS stride (dwords)
constexpr int NSEG      = 3 * KK;         // 3 i-slots x 7 j-planes = 21 segments
constexpr int NITER     = NSEG / 3;       // 7 staging iterations, 3 segments each

// One block per (bc, output row h). Stage the 21 input row-segments this output row can
// touch into LDS with the CDNA5 async copy engine (ASYNCcnt path), then each thread
// gathers <=9 values from LDS, divides by the analytic overlap count, and stores.
__global__ __launch_bounds__(256)
void Patch2Im_fold_kernel(const float* __restrict__ x, float* __restrict__ out) {
    const int h  = blockIdx.x;     // 0..252
    const int bc = blockIdx.y;     // 0..255
    const int t  = threadIdx.x;    // 0..255

    const int h_full = h + PADT;            // 3..255
    const int phbase = h_full / 3;          // 1..85
    const int ri     = h_full - 3 * phbase; // h_full % 3

    __shared__ float smem[NSEG * SEG_STRIDE];

    const float* __restrict__ xbc = x + (size_t)bc * (KK * KK * PLANE);

    // ---- stage phase -----------------------------------------------------------------
    // Segment seg = a*7 + j holds row (phbase-a) of plane (i = ri+3a, j), 85 floats.
    // 255 threads cover 3 segments per iteration: thread t -> (s0 = t/85, off = t%85),
    // iteration it handles segments {3*it, 3*it+1, 3*it+2}.  (/85 hoisted out of loop)
    const int s0  = t / SEG_LEN;            // 0..3 (3 only for t=255 -> inactive)
    const int off = t - s0 * SEG_LEN;       // 0..84
    if (s0 < 3) {
        #pragma unroll
        for (int it = 0; it < NITER; ++it) {
            const int seg = it * 3 + s0;    // 0..20
            const int a   = seg / KK;       // 0..2  (i slot)
            const int j   = seg - a * KK;   // 0..6
            const int i   = ri + 3 * a;
            const int ph  = phbase - a;
            if (i <= 6 && ph >= 0 && ph < NH) {   // block-uniform given (a)
                const float* gsrc = xbc + ((size_t)(i * KK + j) * NH + ph) * NW + off;
                __builtin_amdgcn_global_load_async_to_lds_b32(
                    (AS_GLOBAL int*)(uintptr_t)gsrc,
                    (AS_LDS int*)&smem[seg * SEG_STRIDE + off],
                    /*imm offset*/ 0, /*cpol*/ 0);
            }
        }
    }

    // wait for this wave's async transfers, then barrier so all waves' LDS data is visible
    __builtin_amdgcn_s_wait_asynccnt(0);
    __syncthreads();

    // ---- gather phase: one thread per output column ----------------------------------
    const int w = t;
    if (w < WO) {
        const int w_full = w + PADT;             // 3..255
        const int pwbase = w_full / 3;           // 1..85
        const int rw     = w_full - 3 * pwbase;  // w_full % 3

        // valid j slots (independent of i)
        int joff[3];
        int nj = 0;
        #pragma unroll
        for (int b2 = 0; b2 < 3; ++b2) {
            const int j  = rw + 3 * b2;
            const int pw = pwbase - b2;
            if (j <= 6 && pw >= 0 && pw < NW) {
                joff[nj] = j * SEG_STRIDE + pw;
                ++nj;
            }
        }

        float sum = 0.0f;
        int ni = 0;
        #pragma unroll
        for (int a = 0; a < 3; ++a) {
            const int i  = ri + 3 * a;
            const int ph = phbase - a;
            if (i > 6 || ph < 0 || ph >= NH) continue;
            ++ni;
            const float* srow = &smem[(a * KK) * SEG_STRIDE];
            #pragma unroll
            for (int m = 0; m < 3; ++m) {
                if (m < nj) sum += srow[joff[m]];
            }
        }

        // exact IEEE divide to match the reference's x / counts
        out[((size_t)bc * HO + h) * WO + w] = sum / (float)(ni * nj);
    }
}

extern "C" void kernel_launch(void* const* d_in, const int* in_sizes, int n_in,
                              void* d_out, int out_size, void* d_ws, size_t ws_size,
                              hipStream_t stream) {
    (void)in_sizes; (void)n_in; (void)out_size; (void)d_ws; (void)ws_size;
    const float* x = (const float*)d_in[0];
    float* out     = (float*)d_out;
    dim3 grid(HO, BC, 1);
    dim3 block(256, 1, 1);
    Patch2Im_fold_kernel<<<grid, block, 0, stream>>>(x, out);
}